// Actor_70901320122839
// MI455X (gfx1250) — compile-verified
//
#include <hip/hip_runtime.h>
#include <math.h>

typedef __attribute__((ext_vector_type(2))) float v2f;
typedef __attribute__((ext_vector_type(8))) float v8f;

#define F 128          // feature width (both layers)
#define NGRAPH 64
#define ROWS_PER_WAVE 16
#define WAVES_PER_BLK 10   // 160 rows per block; 100000 % 160 == 0

// ---------------------------------------------------------------------------
// zero fill
__global__ void zero_f(float* __restrict__ p, long n) {
    long i = (long)blockIdx.x * blockDim.x + threadIdx.x;
    long stride = (long)gridDim.x * blockDim.x;
    for (; i < n; i += stride) p[i] = 0.0f;
}

// deg init to 1.0 (self loop)
__global__ void deg_init(float* __restrict__ deg, int n) {
    int i = blockIdx.x * blockDim.x + threadIdx.x;
    if (i < n) deg[i] = 1.0f;
}

__global__ void deg_count(const int* __restrict__ dst, float* __restrict__ deg, int e) {
    int i = blockIdx.x * blockDim.x + threadIdx.x;
    if (i < e) atomicAdd(&deg[dst[i]], 1.0f);
}

__global__ void deg_rsqrt(float* __restrict__ deg, int n) {
    int i = blockIdx.x * blockDim.x + threadIdx.x;
    if (i < n) deg[i] = rsqrtf(deg[i]);   // deg >= 1 always (self loop)
}

// ---------------------------------------------------------------------------
// Y[M,128] = X[M,128] @ W[128,128] using V_WMMA_F32_16X16X4_F32.
// One wave -> 16 rows x 128 cols (8 C-tiles of 16x16). W staged in LDS.
__global__ __launch_bounds__(WAVES_PER_BLK * 32)
void gemm_xw(const float* __restrict__ X, const float* __restrict__ W,
             float* __restrict__ Y, int M) {
    __shared__ float Wlds[F * F];   // 64 KB
    for (int i = threadIdx.x; i < F * F; i += blockDim.x) Wlds[i] = W[i];
    __syncthreads();

    const int wave = threadIdx.x >> 5;
    const int lane = threadIdx.x & 31;
    const int row0 = (blockIdx.x * WAVES_PER_BLK + wave) * ROWS_PER_WAVE;
    const int half = lane >> 4;       // 0 or 1
    const int l16  = lane & 15;
    const int kh   = half * 2;        // K offset within a K=4 step: 0 or 2

    // A row for this lane (clamped for safety; grid is exact for M=100000)
    int arow = row0 + l16; if (arow >= M) arow = M - 1;
    const float* xrow = X + (long)arow * F;

    v8f acc[8] = {};

    for (int k = 0; k < F; k += 4) {
        // A fragment: 16x4 fp32. VGPR0/1 = K=kh, kh+1 ; halves give K 0..3.
        v2f a;
        a.x = xrow[k + kh];
        a.y = xrow[k + kh + 1];
#pragma unroll
        for (int t = 0; t < 8; ++t) {
            // B fragment: 4x16 fp32 tile (cols t*16..t*16+15)
            v2f b;
            b.x = Wlds[(k + kh)     * F + t * 16 + l16];
            b.y = Wlds[(k + kh + 1) * F + t * 16 + l16];
            acc[t] = __builtin_amdgcn_wmma_f32_16x16x4_f32(
                false, a, false, b, (short)0, acc[t], false, false);
        }
    }

    // C/D layout: VGPR r -> row r (lanes 0-15) / row r+8 (lanes 16-31), col = lane&15
    const int rofs = half * 8;
#pragma unroll
    for (int t = 0; t < 8; ++t) {
        float* yb = Y + (long)(row0 + rofs) * F + t * 16 + l16;
#pragma unroll
        for (int r = 0; r < 8; ++r) {
            if (row0 + rofs + r < M) yb[(long)r * F] = acc[t][r];
        }
    }
}

// ---------------------------------------------------------------------------
// One wave per edge: agg[dst] += h[src] * dis[src]*dis[dst]  (float4 per lane)
__global__ void edge_agg(const float* __restrict__ H, const float* __restrict__ dis,
                         const int* __restrict__ src, const int* __restrict__ dst,
                         float* __restrict__ agg, int E) {
    int wid  = (int)(((long)blockIdx.x * blockDim.x + threadIdx.x) >> 5);
    int lane = threadIdx.x & 31;
    if (wid >= E) return;
    int s = src[wid], d = dst[wid];
    float norm = dis[s] * dis[d];
    float4 v = ((const float4*)(H + (long)s * F))[lane];
    float* ad = agg + (long)d * F + lane * 4;
    atomicAdd(ad + 0, v.x * norm);
    atomicAdd(ad + 1, v.y * norm);
    atomicAdd(ad + 2, v.z * norm);
    atomicAdd(ad + 3, v.w * norm);
}

// out = relu(agg + tmp*dis^2 + b)   (self-loop term folded in here)
__global__ void epilogue(const float* __restrict__ tmp, const float* __restrict__ agg,
                         const float* __restrict__ dis, const float* __restrict__ bias,
                         float* __restrict__ out, long total) {
    long i = (long)blockIdx.x * blockDim.x + threadIdx.x;
    if (i >= total) return;
    int node = (int)(i >> 7);
    int c = (int)(i & (F - 1));
    float di = dis[node];
    float v = agg[i] + tmp[i] * di * di + bias[c];
    out[i] = fmaxf(v, 0.0f);
}

// ---------------------------------------------------------------------------
// One wave per node: sums[g] += h[node], counts[g] += 1
__global__ void pool_acc(const float* __restrict__ H, const int* __restrict__ batch,
                         float* __restrict__ sums, float* __restrict__ counts, int N) {
    int wid  = (int)(((long)blockIdx.x * blockDim.x + threadIdx.x) >> 5);
    int lane = threadIdx.x & 31;
    if (wid >= N) return;
    int g = batch[wid];
    float4 v = ((const float4*)(H + (long)wid * F))[lane];
    float* sg = sums + (long)g * F + lane * 4;
    atomicAdd(sg + 0, v.x);
    atomicAdd(sg + 1, v.y);
    atomicAdd(sg + 2, v.z);
    atomicAdd(sg + 3, v.w);
    if (lane == 0) atomicAdd(&counts[g], 1.0f);
}

// Small MLP head: pooled -> relu(fc1) -> sigmoid(fc2)*MAX_ACTION. One block, 128 thr.
__global__ __launch_bounds__(128)
void head(const float* __restrict__ sums, const float* __restrict__ counts,
          const float* __restrict__ fc1w, const float* __restrict__ fc1b,
          const float* __restrict__ fc2w, const float* __restrict__ fc2b,
          float* __restrict__ out) {
    __shared__ float pooled[NGRAPH * F];   // 32 KB
    __shared__ float z[NGRAPH * F];        // 32 KB
    int tid = threadIdx.x;
    for (int g = 0; g < NGRAPH; ++g)
        pooled[g * F + tid] = sums[g * F + tid] / fmaxf(counts[g], 1.0f);
    __syncthreads();
    for (int g = 0; g < NGRAPH; ++g) {
        float acc = fc1b[tid];
        for (int k = 0; k < F; ++k) acc += pooled[g * F + k] * fc1w[k * F + tid];
        z[g * F + tid] = fmaxf(acc, 0.0f);
    }
    __syncthreads();
    for (int idx = tid; idx < NGRAPH * 8; idx += 128) {
        int g = idx >> 3, a = idx & 7;
        float acc = fc2b[a];
        for (int k = 0; k < F; ++k) acc += z[g * F + k] * fc2w[k * 8 + a];
        out[idx] = 1.0f / (1.0f + expf(-acc));   // * MAX_ACTION (=1.0)
    }
}

// ---------------------------------------------------------------------------
extern "C" void kernel_launch(void* const* d_in, const int* in_sizes, int n_in,
                              void* d_out, int out_size, void* d_ws, size_t ws_size,
                              hipStream_t stream) {
    const float* x     = (const float*)d_in[0];
    const int*   eidx  = (const int*)d_in[1];
    const int*   batch = (const int*)d_in[2];
    const float* W1    = (const float*)d_in[3];
    const float* b1    = (const float*)d_in[4];
    const float* W2    = (const float*)d_in[5];
    const float* b2    = (const float*)d_in[6];
    const float* fc1w  = (const float*)d_in[7];
    const float* fc1b  = (const float*)d_in[8];
    const float* fc2w  = (const float*)d_in[9];
    const float* fc2b  = (const float*)d_in[10];

    const int N = in_sizes[0] / F;       // 100000
    const int E = in_sizes[1] / 2;       // 1600000
    const int* src = eidx;
    const int* dst = eidx + E;
    const long NF = (long)N * F;         // 12.8M

    float* ws     = (float*)d_ws;
    float* tmp    = ws;                  // [N,128]  X@W
    float* agg    = tmp + NF;            // [N,128]  edge accumulator
    float* h      = agg + NF;            // [N,128]  layer output
    float* dis    = h + NF;              // [N]      D^-1/2
    float* sums   = dis + N;             // [64,128]
    float* counts = sums + NGRAPH * F;   // [64]

    const int T = 256;
    dim3 tBlk(T);
    int nBlkN   = (N + T - 1) / T;
    int nBlkE   = (E + T - 1) / T;
    int nBlkNF  = (int)((NF + T - 1) / T);
    int gemmBlk = (N + ROWS_PER_WAVE * WAVES_PER_BLK - 1) / (ROWS_PER_WAVE * WAVES_PER_BLK);
    int edgeWBlk = (E * 32 + T - 1) / T;   // one wave per edge
    int poolWBlk = (N * 32 + T - 1) / T;   // one wave per node

    // degrees -> dis
    deg_init<<<nBlkN, tBlk, 0, stream>>>(dis, N);
    deg_count<<<nBlkE, tBlk, 0, stream>>>(dst, dis, E);
    deg_rsqrt<<<nBlkN, tBlk, 0, stream>>>(dis, N);

    // ---- layer 1 ----
    zero_f<<<nBlkNF, tBlk, 0, stream>>>(agg, NF);
    gemm_xw<<<gemmBlk, WAVES_PER_BLK * 32, 0, stream>>>(x, W1, tmp, N);
    edge_agg<<<edgeWBlk, tBlk, 0, stream>>>(tmp, dis, src, dst, agg, E);
    epilogue<<<nBlkNF, tBlk, 0, stream>>>(tmp, agg, dis, b1, h, NF);

    // ---- layer 2 ----
    zero_f<<<nBlkNF, tBlk, 0, stream>>>(agg, NF);
    gemm_xw<<<gemmBlk, WAVES_PER_BLK * 32, 0, stream>>>(h, W2, tmp, N);
    edge_agg<<<edgeWBlk, tBlk, 0, stream>>>(tmp, dis, src, dst, agg, E);
    epilogue<<<nBlkNF, tBlk, 0, stream>>>(tmp, agg, dis, b2, h, NF);

    // ---- pooling + head ----
    zero_f<<<(NGRAPH * F + NGRAPH + T - 1) / T, tBlk, 0, stream>>>(sums, NGRAPH * F + NGRAPH);
    pool_acc<<<poolWBlk, tBlk, 0, stream>>>(h, batch, sums, counts, N);
    head<<<1, 128, 0, stream>>>(sums, counts, fc1w, fc1b, fc2w, fc2b, (float*)d_out);
}